// Self_Attn_45243185496183
// MI455X (gfx1250) — compile-verified
//
#include <hip/hip_runtime.h>

typedef float v2f __attribute__((ext_vector_type(2)));
typedef float v8f __attribute__((ext_vector_type(8)));

// ---- problem sizes ----
constexpr int B_ = 8, C_ = 128, Hh = 64, Ww = 64;
constexpr int N_ = Hh * Ww;      // 4096
constexpr int M_ = N_ / 4;       // 1024 (after 2x2 pool)
constexpr int C8_ = 16, C2_ = 64;

// ---- attention kernel tiling ----
constexpr int TN = 64;           // n-rows per workgroup
constexpr int SLD = 1026;        // S row stride (even -> 8B-aligned v2f; 1026%64==2 -> banks 2l/2l+1)
constexpr int PHI_MC = 512;      // phi chunk width
constexpr int G_MC = 64;         // g chunk width (32 K-pairs)
constexpr int GPAD = 66;         // pair-row pad for g chunk
constexpr int OPAD = 66;         // row pad for O  [n][c]
constexpr int WAPAD = 66;        // row pad for staged w_attn [o][c]

constexpr size_t S_FLOATS   = (size_t)TN * SLD;   // 65664 (reused for w_attn stage: 128*66=8448)
constexpr size_t TH_FLOATS  = 8 * 64 * 2;         // 1024
constexpr size_t BUF_FLOATS = 8 * PHI_MC * 2;     // 8192
constexpr size_t LDS_BYTES  = (S_FLOATS + TH_FLOATS + BUF_FLOATS) * 4;  // 299520 < 320KB

__device__ __forceinline__ v8f wmma_f32(v2f a, v2f b, v8f c) {
  return __builtin_amdgcn_wmma_f32_16x16x4_f32(false, a, false, b, (short)0, c, false, false);
}
__device__ __forceinline__ v2f lds2(const float* p) {  // one ds_load_b64
  return *reinterpret_cast<const v2f*>(p);
}

// ------------------------------------------------------------------
// Kernel 1: theta = w_theta @ x + b_theta      [B,16,N]
// ------------------------------------------------------------------
__global__ __launch_bounds__(256) void theta_proj_kernel(
    const float* __restrict__ x, const float* __restrict__ w,
    const float* __restrict__ bias, float* __restrict__ theta) {
  __shared__ float ws[C8_ * C_];
  __shared__ float bs[C8_];
  const int t = threadIdx.x;
  for (int i = t; i < C8_ * C_; i += 256) ws[i] = w[i];
  if (t < C8_) bs[t] = bias[t];
  __syncthreads();

  const int b = blockIdx.x >> 4;
  const int p = ((blockIdx.x & 15) << 8) + t;
  float acc[C8_];
#pragma unroll
  for (int d = 0; d < C8_; ++d) acc[d] = 0.f;
  for (int c = 0; c < C_; ++c) {
    const float xv = x[((size_t)(b * C_ + c)) * N_ + p];
#pragma unroll
    for (int d = 0; d < C8_; ++d) acc[d] = fmaf(ws[d * C_ + c], xv, acc[d]);
  }
#pragma unroll
  for (int d = 0; d < C8_; ++d)
    theta[((size_t)(b * C8_ + d)) * N_ + p] = acc[d] + bs[d];
}

// ------------------------------------------------------------------
// Kernel 2: phi = pool(w_phi@x+b), g = pool(w_g@x+b)
// ------------------------------------------------------------------
__global__ __launch_bounds__(256) void pool_proj_kernel(
    const float* __restrict__ x,
    const float* __restrict__ w_phi, const float* __restrict__ b_phi,
    const float* __restrict__ w_g,   const float* __restrict__ b_g,
    float* __restrict__ phi, float* __restrict__ g) {
  const int tid = blockIdx.x * 256 + threadIdx.x;
  const int m  = tid & (M_ - 1);
  const int r  = tid >> 10;
  const int b  = r & 7;
  const int ch = r >> 3;
  const int ph = m >> 5, pw = m & 31;
  const int p00 = (ph * 2) * Ww + pw * 2;

  const float* wrow = (ch < C8_) ? (w_phi + ch * C_) : (w_g + (ch - C8_) * C_);
  const float  bias = (ch < C8_) ? b_phi[ch] : b_g[ch - C8_];
  float a0 = bias, a1 = bias, a2 = bias, a3 = bias;
  const float* xb = x + (size_t)b * C_ * N_;
  for (int c = 0; c < C_; ++c) {
    const float wv = wrow[c];
    const float* xp = xb + (size_t)c * N_ + p00;
    a0 = fmaf(wv, xp[0], a0);
    a1 = fmaf(wv, xp[1], a1);
    a2 = fmaf(wv, xp[Ww], a2);
    a3 = fmaf(wv, xp[Ww + 1], a3);
  }
  const float mx = fmaxf(fmaxf(a0, a1), fmaxf(a2, a3));
  if (ch < C8_) phi[((size_t)b * C8_ + ch) * M_ + m] = mx;
  else          g[((size_t)b * C2_ + (ch - C8_)) * M_ + m] = mx;
}

// ------------------------------------------------------------------
// Kernel 3: fused attention; batched ds_load_b64 fragment groups feed
// v_wmma_f32_16x16x4_f32 chains (independent regs -> loads pipeline).
// ------------------------------------------------------------------
__global__ __launch_bounds__(256) void attn_fused_kernel(
    const float* __restrict__ x, const float* __restrict__ theta,
    const float* __restrict__ phi, const float* __restrict__ g,
    const float* __restrict__ w_attn, const float* __restrict__ b_attn,
    const float* __restrict__ sigma, float* __restrict__ out) {
  extern __shared__ float smem[];
  float* S   = smem;                       // [64][SLD]  scores
  float* WA  = smem;                       // [128][66]  staged w_attn (phase C; S dead)
  float* TH  = smem + S_FLOATS;            // theta: [kp(8)][n(64)][2]
  float* BUF = TH + TH_FLOATS;             // phi pairs / g pairs / O

  const int t    = threadIdx.x;
  const int wv   = t >> 5;
  const int lane = t & 31;
  const int half = lane >> 4;
  const int l    = lane & 15;

  const int b  = blockIdx.x >> 6;
  const int n0 = (blockIdx.x & 63) * TN;

  const float* thetaB = theta + (size_t)b * C8_ * N_;
  const float* phiB   = phi   + (size_t)b * C8_ * M_;
  const float* gB     = g     + (size_t)b * C2_ * M_;

  for (int i = t; i < C8_ * TN; i += 256) {
    int d = i >> 6, j = i & 63;
    TH[(((d >> 1) * 64) + j) * 2 + (d & 1)] = thetaB[d * N_ + n0 + j];
  }

  // ---------- Phase A: S[64][1024] = theta^T @ phi ----------
  for (int mc = 0; mc < M_ / PHI_MC; ++mc) {
    __syncthreads();
    for (int i = t; i < C8_ * PHI_MC; i += 256) {
      int d = i >> 9, m = i & (PHI_MC - 1);
      BUF[(((d >> 1) * PHI_MC) + m) * 2 + (d & 1)] = phiB[d * M_ + mc * PHI_MC + m];
    }
    if (mc + 1 < M_ / PHI_MC) {
      __builtin_prefetch(phiB + (t & 15) * M_ + (mc + 1) * PHI_MC + (t >> 4) * 32, 0, 1);
    }
    __syncthreads();
    for (int nb = 0; nb < 4; ++nb) {
      const int nbase = nb * 16;
      v2f afr[4];
#pragma unroll
      for (int i = 0; i < 4; ++i)
        afr[i] = lds2(TH + (((2 * i + half) * 64) + nbase + l) * 2);
#pragma unroll
      for (int j = 0; j < 4; ++j) {
        const int mlb = (wv + 8 * j) * 16;
        v2f bb[4];                                   // batch loads: independent regs
#pragma unroll
        for (int i = 0; i < 4; ++i)
          bb[i] = lds2(BUF + (((2 * i + half) * PHI_MC) + mlb + l) * 2);
        v8f acc = {};
#pragma unroll
        for (int i = 0; i < 4; ++i) acc = wmma_f32(afr[i], bb[i], acc);
        float* srow = S + (nbase + 8 * half) * SLD + mc * PHI_MC + mlb + l;
#pragma unroll
        for (int r = 0; r < 8; ++r) srow[r * SLD] = acc[r];
      }
    }
  }
  __syncthreads();

  // ---------- softmax rows (wave w owns rows 8w..8w+7) ----------
  for (int rr = 0; rr < 8; ++rr) {
    const int row = wv * 8 + rr;
    float vals[32];
    float mx = -3.402823e38f;
#pragma unroll
    for (int j = 0; j < 32; ++j) {
      vals[j] = S[row * SLD + lane + 32 * j];
      mx = fmaxf(mx, vals[j]);
    }
    for (int off = 16; off > 0; off >>= 1) mx = fmaxf(mx, __shfl_xor(mx, off, 32));
    float sum = 0.f;
#pragma unroll
    for (int j = 0; j < 32; ++j) { vals[j] = __expf(vals[j] - mx); sum += vals[j]; }
    for (int off = 16; off > 0; off >>= 1) sum += __shfl_xor(sum, off, 32);
    const float inv = 1.0f / sum;
#pragma unroll
    for (int j = 0; j < 32; ++j) S[row * SLD + lane + 32 * j] = vals[j] * inv;
  }

  // ---------- Phase B: O[64 n][64 c] = S @ g^T ----------
  const int nbB = (wv >> 1) * 16;
  const int cb0 = (wv & 1) * 32;
  const float* sArow = S + (nbB + l) * SLD + 2 * half;
  v8f o0 = {}, o1 = {};
  for (int mc = 0; mc < M_ / G_MC; ++mc) {
    __syncthreads();
    for (int i = t; i < C2_ * G_MC; i += 256) {
      int c = i >> 6, ml = i & 63;
      BUF[(((ml >> 1) * GPAD) + c) * 2 + (ml & 1)] = gB[c * M_ + mc * G_MC + ml];
    }
    if (mc + 1 < M_ / G_MC && t < 128) {
      __builtin_prefetch(gB + (t & 63) * M_ + (mc + 1) * G_MC + (t >> 6) * 32, 0, 1);
    }
    __syncthreads();
#pragma unroll 2
    for (int kk = 0; kk < G_MC; kk += 8) {           // batch 6 loads -> 4 wmma
      const int mp0 = (kk >> 1) + half;
      const int mp1 = mp0 + 2;
      const v2f a0  = lds2(sArow + mc * G_MC + kk);
      const v2f a1  = lds2(sArow + mc * G_MC + kk + 4);
      const v2f b00 = lds2(BUF + ((mp0 * GPAD) + cb0 + l) * 2);
      const v2f b01 = lds2(BUF + ((mp0 * GPAD) + cb0 + 16 + l) * 2);
      const v2f b10 = lds2(BUF + ((mp1 * GPAD) + cb0 + l) * 2);
      const v2f b11 = lds2(BUF + ((mp1 * GPAD) + cb0 + 16 + l) * 2);
      o0 = wmma_f32(a0, b00, o0);
      o1 = wmma_f32(a0, b01, o1);
      o0 = wmma_f32(a1, b10, o0);
      o1 = wmma_f32(a1, b11, o1);
    }
  }
  __syncthreads();

  // store O -> BUF [n][c] pad 66; stage w_attn -> dead S region [o][c] pad 66
#pragma unroll
  for (int r = 0; r < 8; ++r) {
    BUF[(nbB + r + 8 * half) * OPAD + cb0 + l]      = o0[r];
    BUF[(nbB + r + 8 * half) * OPAD + cb0 + 16 + l] = o1[r];
  }
  for (int i = t; i < C_ * C2_; i += 256) {
    int o = i >> 6, c = i & 63;
    WA[o * WAPAD + c] = w_attn[o * C2_ + c];
  }
  __syncthreads();

  // ---------- Phase C: Y[128 o][64 n] = w_attn @ O^T ; residual writeout ----------
  const float sg = sigma[0];
  const int nbC = (wv >> 1) * 16;
  const float* obrow = BUF + (nbC + l) * OPAD + 2 * half;
  v8f acc[4] = {v8f{}, v8f{}, v8f{}, v8f{}};
#pragma unroll 4
  for (int kk = 0; kk < C2_; kk += 4) {
    const v2f bb = lds2(obrow + kk);
    v2f af[4];                                       // batch the 4 A-frags
#pragma unroll
    for (int ti = 0; ti < 4; ++ti)
      af[ti] = lds2(WA + ((((wv & 1) * 4 + ti) * 16) + l) * WAPAD + kk + 2 * half);
#pragma unroll
    for (int ti = 0; ti < 4; ++ti) acc[ti] = wmma_f32(af[ti], bb, acc[ti]);
  }
#pragma unroll
  for (int ti = 0; ti < 4; ++ti) {
    const int ob = ((wv & 1) * 4 + ti) * 16;
#pragma unroll
    for (int r = 0; r < 8; ++r) {
      const int o = ob + r + 8 * half;
      const int n = n0 + nbC + l;
      const size_t idx = ((size_t)b * C_ + o) * N_ + n;
      out[idx] = x[idx] + sg * (acc[ti][r] + b_attn[o]);
    }
  }
}

// ------------------------------------------------------------------
extern "C" void kernel_launch(void* const* d_in, const int* in_sizes, int n_in,
                              void* d_out, int out_size, void* d_ws, size_t ws_size,
                              hipStream_t stream) {
  (void)in_sizes; (void)n_in; (void)out_size; (void)ws_size;
  const float* x       = (const float*)d_in[0];
  const float* w_theta = (const float*)d_in[1];
  const float* b_theta = (const float*)d_in[2];
  const float* w_phi   = (const float*)d_in[3];
  const float* b_phi   = (const float*)d_in[4];
  const float* w_g     = (const float*)d_in[5];
  const float* b_g     = (const float*)d_in[6];
  const float* w_attn  = (const float*)d_in[7];
  const float* b_attn  = (const float*)d_in[8];
  const float* sigma   = (const float*)d_in[9];
  float* out = (float*)d_out;

  float* theta_ws = (float*)d_ws;
  float* phi_ws   = theta_ws + (size_t)B_ * C8_ * N_;
  float* g_ws     = phi_ws   + (size_t)B_ * C8_ * M_;

  theta_proj_kernel<<<B_ * (N_ / 256), 256, 0, stream>>>(x, w_theta, b_theta, theta_ws);
  pool_proj_kernel<<<(B_ * M_ * (C8_ + C2_)) / 256, 256, 0, stream>>>(
      x, w_phi, b_phi, w_g, b_g, phi_ws, g_ws);
  attn_fused_kernel<<<B_ * (N_ / TN), 256, (unsigned)LDS_BYTES, stream>>>(
      x, theta_ws, phi_ws, g_ws, w_attn, b_attn, sigma, out);
}